// MixedBoundaryGaussian_BoundaryDoU_68032281969167
// MI455X (gfx1250) — compile-verified
//
#include <hip/hip_runtime.h>
#include <math.h>

#define B_ 16
#define H_ 384
#define W_ 384
#define N_ (B_ * H_ * W_)        // 2359296
#define NROWS (B_ * H_)          // 6144
#define NCOLS (B_ * W_)          // 6144
#define STAT_BLOCKS 1152
#define STAT_TPB 256
#define STAT_EPT 8               // 1152*256*8 == N_ exactly

typedef __attribute__((ext_vector_type(2))) float v2f;
typedef __attribute__((ext_vector_type(8))) float v8f;

// ---------------------------------------------------------------------------
// Exact f32 wave32 sum through the matrix pipe:
// A = (x, 0) per lane (16x4 f32 A-matrix), B = all ones (4x16) -> D[m][n] =
// rowsum(m). Each lane's 8 accumulator VGPRs cover 8 distinct rows (one half
// of 16); lane^16 covers the other half, so sum(d[0..7]) + xor-16 exchange
// == full wave sum. ds_swizzle group-of-32: offset = and_mask(0x1f) |
// (or_mask 0)<<5 | (xor_mask 0x10)<<10 = 0x401f. EXEC all-ones at call sites.
// ---------------------------------------------------------------------------
__device__ __forceinline__ float wave_sum32(float x) {
  v2f a; a.x = x;    a.y = 0.0f;
  v2f b; b.x = 1.0f; b.y = 1.0f;
  v8f c = {};
  v8f d = __builtin_amdgcn_wmma_f32_16x16x4_f32(false, a, false, b, (short)0, c,
                                                false, false);
  float s = d[0] + d[1] + d[2] + d[3] + d[4] + d[5] + d[6] + d[7];
  int other = __builtin_amdgcn_ds_swizzle(__float_as_int(s), 0x401f);
  s += __int_as_float(other);
  return s;
}

// ---------------------------------------------------------------------------
// Pass 1: vertical two-scan distance (exact replica of reference scans),
// writes f1 = d1*d1. One thread per (b, w) column; coalesced along w.
// ---------------------------------------------------------------------------
__global__ void k_col(const float* __restrict__ targets, float* __restrict__ f1) {
  int t = blockIdx.x * blockDim.x + threadIdx.x;   // 0 .. NCOLS-1
  int b = t / W_;
  int w = t - b * W_;
  const float* tg = targets + (size_t)b * H_ * W_ + w;
  float* out = f1 + (size_t)b * H_ * W_ + w;

  float c = 10000.0f;
  for (int h = 0; h < H_; ++h) {
    float m = tg[(size_t)h * W_];
    c = (m > 0.5f) ? 0.0f : (c + 1.0f);
    out[(size_t)h * W_] = c;
  }
  c = 10000.0f;
  for (int h = H_ - 1; h >= 0; --h) {
    float m = tg[(size_t)h * W_];
    c = (m > 0.5f) ? 0.0f : (c + 1.0f);
    float d = fminf(out[(size_t)h * W_], c);
    out[(size_t)h * W_] = d * d;
  }
}

// ---------------------------------------------------------------------------
// Pass 2: global stats. 7 sums: S_pos, C1, C0, I1, I0, z1, z0.
// Cross-stencil with zero padding, per image (guards on h/w).
// ---------------------------------------------------------------------------
__global__ void k_stats(const float* __restrict__ probs,
                        const float* __restrict__ targets,
                        float* __restrict__ partials) {
  float q0 = 0.f, q1 = 0.f, q2 = 0.f, q3 = 0.f, q4 = 0.f, q5 = 0.f, q6 = 0.f;
  int base = blockIdx.x * STAT_TPB + threadIdx.x;
  for (int e = 0; e < STAT_EPT; ++e) {
    int i = base + e * (STAT_BLOCKS * STAT_TPB);
    int w = i % W_;
    int h = (i / W_) % H_;
    float y  = (targets[i] > 0.5f) ? 1.0f : 0.0f;
    float pc = fminf(fmaxf(probs[i], 1e-7f), 1.0f - 1e-7f);
    float c1 = y;            // conv of (t==1) with zero pad
    float c0 = 1.0f - y;     // conv of (t==0) with zero pad
    if (h > 0)      { float yn = (targets[i - W_] > 0.5f) ? 1.f : 0.f; c1 += yn; c0 += 1.f - yn; }
    if (h < H_ - 1) { float yn = (targets[i + W_] > 0.5f) ? 1.f : 0.f; c1 += yn; c0 += 1.f - yn; }
    if (w > 0)      { float yn = (targets[i - 1 ] > 0.5f) ? 1.f : 0.f; c1 += yn; c0 += 1.f - yn; }
    if (w < W_ - 1) { float yn = (targets[i + 1 ] > 0.5f) ? 1.f : 0.f; c1 += yn; c0 += 1.f - yn; }
    q0 += y;
    q1 += (y != 0.0f && c1 != 5.0f) ? 1.0f : 0.0f;   // C1
    q2 += (y == 0.0f && c0 != 5.0f) ? 1.0f : 0.0f;   // C0
    q3 += pc * y;                                    // intersect (t==1)
    q4 += (1.0f - pc) * (1.0f - y);                  // intersect (t==0)
    q5 += pc * pc;                                   // z_sum (t==1 call)
    q6 += (1.0f - pc) * (1.0f - pc);                 // z_sum (t==0 call)
  }

  __shared__ float lds[7][8];
  int lane = threadIdx.x & 31, wid = threadIdx.x >> 5;
  float qs[7] = {q0, q1, q2, q3, q4, q5, q6};
  for (int k = 0; k < 7; ++k) {
    float wsum = wave_sum32(qs[k]);   // all 256 threads active -> EXEC all ones
    if (lane == 0) lds[k][wid] = wsum;
  }
  __syncthreads();
  if (threadIdx.x == 0) {
    for (int k = 0; k < 7; ++k) {
      float s = 0.0f;
      for (int w2 = 0; w2 < 8; ++w2) s += lds[k][w2];
      partials[blockIdx.x * 8 + k] = s;
    }
  }
}

// ---------------------------------------------------------------------------
// Pass 3: fold stat partials, compute scalars: w_pos, w_neg_base, dou loss.
// ---------------------------------------------------------------------------
__global__ void k_scalars(const float* __restrict__ partials, float* __restrict__ scal) {
  __shared__ float red[256];
  __shared__ float qs[8];
  int tid = threadIdx.x;
  int q = tid & 7;
  int slot = tid >> 3;                 // 0..31
  float s = 0.0f;
  for (int j = slot; j < STAT_BLOCKS; j += 32) s += partials[j * 8 + q];
  red[tid] = s;
  __syncthreads();
  if (tid < 8) {
    float t = 0.0f;
    for (int sl = 0; sl < 32; ++sl) t += red[tid + 8 * sl];
    qs[tid] = t;
  }
  __syncthreads();
  if (tid == 0) {
    const float eps = 1e-6f, smooth = 1e-5f;
    float Spos = qs[0], C1 = qs[1], C0 = qs[2], I1 = qs[3], I0 = qs[4],
          z1 = qs[5], z0 = qs[6];
    float Sneg = (float)N_ - Spos;
    float w_pos = fminf((Sneg + eps) / (Spos + eps), 1.0f);
    float w_neg_base = (Spos + eps) / (Sneg + eps);
    float a1 = fminf(2.0f * (1.0f - (C1 + smooth) / (Spos + smooth)) - 1.0f, 0.8f);
    float a0 = fminf(2.0f * (1.0f - (C0 + smooth) / (Sneg + smooth)) - 1.0f, 0.8f);
    float dou1 = (z1 + Spos - 2.0f * I1 + smooth) /
                 (z1 + Spos - (1.0f + a1) * I1 + smooth);
    float dou0 = (z0 + Sneg - 2.0f * I0 + smooth) /
                 (z0 + Sneg - (1.0f + a0) * I0 + smooth);
    scal[0] = w_pos;
    scal[1] = w_neg_base;
    scal[2] = 0.5f * (dou0 + dou1);
  }
}

// ---------------------------------------------------------------------------
// Pass 4: per-row exact column envelope (min_k f1[k] + (w-k)^2) + bg loss term.
// One block per (b,h) row, 384 threads (12 waves, full EXEC).
// Row staged into LDS via the gfx1250 async global->LDS path when available
// (ASYNCcnt-tracked; each wave waits its own asynccnt before the barrier).
// ---------------------------------------------------------------------------
__global__ void k_row(const float* __restrict__ probs,
                      const float* __restrict__ targets,
                      const float* __restrict__ f1,
                      const float* __restrict__ scal,
                      float* __restrict__ rowp) {
  __shared__ float fr[W_];
  __shared__ float lds[12];
  int r = blockIdx.x;                  // b*H + h
  int w = threadIdx.x;
  size_t rowoff = (size_t)r * W_;

#if __has_builtin(__builtin_amdgcn_global_load_async_to_lds_b32)
  __builtin_amdgcn_global_load_async_to_lds_b32(
      (__attribute__((address_space(1))) int*)(f1 + rowoff + w),
      (__attribute__((address_space(3))) int*)&fr[w],
      /*offset=*/0, /*cpol=*/0);
#if __has_builtin(__builtin_amdgcn_s_wait_asynccnt)
  __builtin_amdgcn_s_wait_asynccnt(0);
#else
  asm volatile("s_wait_asynccnt 0x0" ::: "memory");
#endif
#else
  fr[w] = f1[rowoff + w];
#endif
  __syncthreads();

  float best = 1e10f;
  float fw = (float)w;
#pragma unroll 4
  for (int k = 0; k < W_; ++k) {
    float dk = fw - (float)k;
    best = fminf(best, fmaf(dk, dk, fr[k]));
  }
  float wb = expf(-best * 0.125f);     // 2*sigma^2 = 8

  float y = (targets[rowoff + w] > 0.5f) ? 1.0f : 0.0f;
  float p = fminf(fmaxf(probs[rowoff + w], 1e-6f), 1.0f - 1e-6f);
  float w_pos = scal[0];
  float w_neg = fminf(scal[1] + wb, 1.0f);
  float contrib = w_neg * (1.0f - y) * (-log1pf(-p)) + w_pos * y * (-logf(p));

  float s = wave_sum32(contrib);
  int lane = threadIdx.x & 31, wid = threadIdx.x >> 5;
  if (lane == 0) lds[wid] = s;
  __syncthreads();
  if (threadIdx.x == 0) {
    float tot = 0.0f;
    for (int i = 0; i < 12; ++i) tot += lds[i];
    rowp[r] = tot;
  }
}

// ---------------------------------------------------------------------------
// Pass 5: fold row partials -> final scalar.
// ---------------------------------------------------------------------------
__global__ void k_final(const float* __restrict__ rowp,
                        const float* __restrict__ scal,
                        float* __restrict__ out) {
  __shared__ float red[256];
  int tid = threadIdx.x;
  float s = 0.0f;
  for (int j = tid; j < NROWS; j += 256) s += rowp[j];
  red[tid] = s;
  __syncthreads();
  if (tid == 0) {
    float tot = 0.0f;
    for (int i = 0; i < 256; ++i) tot += red[i];
    out[0] = tot / (float)N_ + scal[2];   // LAMBDA_DOU = 1
  }
}

extern "C" void kernel_launch(void* const* d_in, const int* in_sizes, int n_in,
                              void* d_out, int out_size, void* d_ws, size_t ws_size,
                              hipStream_t stream) {
  const float* probs   = (const float*)d_in[0];
  const float* targets = (const float*)d_in[1];
  float* out = (float*)d_out;

  float* ws = (float*)d_ws;
  float* f1       = ws;                          // N_ floats
  float* partials = f1 + N_;                     // STAT_BLOCKS*8 floats
  float* scal     = partials + STAT_BLOCKS * 8;  // 8 floats
  float* rowp     = scal + 8;                    // NROWS floats

  k_col<<<NCOLS / 256, 256, 0, stream>>>(targets, f1);
  k_stats<<<STAT_BLOCKS, STAT_TPB, 0, stream>>>(probs, targets, partials);
  k_scalars<<<1, 256, 0, stream>>>(partials, scal);
  k_row<<<NROWS, W_, 0, stream>>>(probs, targets, f1, scal, rowp);
  k_final<<<1, 256, 0, stream>>>(rowp, scal, out);
}